// Smooth_JPEG_36206574305948
// MI455X (gfx1250) — compile-verified
//
#include <hip/hip_runtime.h>

typedef __attribute__((ext_vector_type(2))) float v2f;
typedef __attribute__((ext_vector_type(8))) float v8f;

#define B_ 4
#define H_ 512
#define W_ 512
#define PI_ 3.14159265358979323846f

static const int NIMG  = B_ * H_ * W_ * 3;     // 3,145,728 floats
static const int NYC   = B_ * 64 * 64 * 64;    // 1,048,576
static const int NUC   = B_ * 32 * 32 * 64;    //   262,144
static const int NCOEF = NYC + 2 * NUC;        // 1,572,864

// JPEG luminance quant table / 255
__constant__ float c_qt[64] = {
  16.f/255, 11.f/255, 10.f/255, 16.f/255, 24.f/255, 40.f/255, 51.f/255, 61.f/255,
  12.f/255, 12.f/255, 14.f/255, 19.f/255, 26.f/255, 58.f/255, 60.f/255, 55.f/255,
  14.f/255, 13.f/255, 16.f/255, 24.f/255, 40.f/255, 57.f/255, 69.f/255, 56.f/255,
  14.f/255, 17.f/255, 22.f/255, 29.f/255, 51.f/255, 87.f/255, 80.f/255, 62.f/255,
  18.f/255, 22.f/255, 37.f/255, 56.f/255, 68.f/255,109.f/255,103.f/255, 77.f/255,
  24.f/255, 35.f/255, 55.f/255, 64.f/255, 81.f/255,104.f/255,113.f/255, 92.f/255,
  49.f/255, 64.f/255, 78.f/255, 87.f/255,103.f/255,121.f/255,120.f/255,101.f/255,
  72.f/255, 92.f/255, 95.f/255, 98.f/255,112.f/255,100.f/255,103.f/255, 99.f/255 };

// Build the full 16x16 block-diagonal DCT matrix M = blockdiag(D8,D8) in LDS,
// zeros included, so all fragment reads are unconditional ds_loads (no EXEC
// predication around LDS traffic). One element per thread; blockDim.x == 256.
__device__ __forceinline__ void init_M16(float* M16, int tid) {
  int x = tid >> 4, u = tid & 15;
  float v = 0.0f;
  if ((x >> 3) == (u >> 3)) {
    int xl = x & 7, ul = u & 7;
    float cu = (ul == 0) ? 0.35355339059327373f : 0.5f;   // 0.5*a(u), a(0)=1/sqrt2
    v = cu * cosf((2.0f * (float)xl + 1.0f) * (float)ul * PI_ / 16.0f);
  }
  M16[tid] = v;   // row-major: M16[x*16 + u]
}

__device__ __forceinline__ v8f wmma_acc(v2f a, v2f b, v8f c) {
  // D = A(16x4,f32) * B(4x16,f32) + C(16x16,f32)
  return __builtin_amdgcn_wmma_f32_16x16x4_f32(false, a, false, b, (short)0, c,
                                               false, false);
}

__device__ __forceinline__ float clampf(float v, float lo, float hi) {
  return fminf(fmaxf(v, lo), hi);
}

// ---------------------------------------------------------------------------
// Forward W: RGB image -> DCT coefs for Y (full-res) and U,V (2x downsampled).
// One wave per 16x16 channel tile (= four 8x8 blocks). Jobs:
//   [0,4096)      : Y   tiles (4 batches x 32x32 tiles)
//   [4096,5120)   : U   tiles (4 batches x 16x16 tiles of 256x256 plane)
//   [5120,6144)   : V   tiles
// ---------------------------------------------------------------------------
__global__ __launch_bounds__(256) void k_forward_W(const float* __restrict__ img,
                                                   float* __restrict__ yc,
                                                   float* __restrict__ uc,
                                                   float* __restrict__ vc) {
  __shared__ float M16[256];
  __shared__ float S[8][256];   // per-wave staging (X tile, then T)
  int tid = threadIdx.x;
  init_M16(M16, tid);
  __syncthreads();

  int wave = tid >> 5, lane = tid & 31;
  int job = blockIdx.x * 8 + wave;

  int ch, b, th, tw;
  if (job < 4096)      { ch = 0; b = job >> 10; int t = job & 1023; th = t >> 5; tw = t & 31; }
  else if (job < 5120) { ch = 1; int t = job - 4096; b = t >> 8; th = (t >> 4) & 15; tw = t & 15; }
  else                 { ch = 2; int t = job - 5120; b = t >> 8; th = (t >> 4) & 15; tw = t & 15; }

  float* X = S[wave];
  // Stage the 16x16 channel tile (color convert / downsample fused here)
  for (int e = lane; e < 256; e += 32) {
    int i = e >> 4, j = e & 15;
    float val;
    if (ch == 0) {
      int py = th * 16 + i, px = tw * 16 + j;
      const float* p = img + (((size_t)b * H_ + py) * W_ + px) * 3;
      val = 0.299f * p[0] + 0.587f * p[1] + 0.114f * p[2] - 0.5f;
    } else {
      int py = (th * 16 + i) * 2, px = (tw * 16 + j) * 2;
      const float* p0 = img + (((size_t)b * H_ + py) * W_ + px) * 3;
      const float* p1 = p0 + (size_t)W_ * 3;
      float r  = 0.25f * (p0[0] + p0[3] + p1[0] + p1[3]);
      float g  = 0.25f * (p0[1] + p0[4] + p1[1] + p1[4]);
      float bl = 0.25f * (p0[2] + p0[5] + p1[2] + p1[5]);
      val = (ch == 1) ? (-0.14714119f * r - 0.28886916f * g + 0.43601035f * bl)
                      : ( 0.61497538f * r - 0.51496512f * g - 0.10001026f * bl);
    }
    X[i * 16 + j] = val;
  }

  int mA = lane & 15, nB = lane & 15, koff = (lane >> 4) * 2;

  // step 1: T = M^T * X
  v8f acc = {0, 0, 0, 0, 0, 0, 0, 0};
#pragma unroll
  for (int c = 0; c < 4; ++c) {
    int kb = 4 * c + koff;
    v2f a, bb;
    a.x  = M16[(kb + 0) * 16 + mA];   // A[m][k] = M[k][m]
    a.y  = M16[(kb + 1) * 16 + mA];
    bb.x = X[(kb + 0) * 16 + nB];     // B[k][n] = X[k][n]
    bb.y = X[(kb + 1) * 16 + nB];
    acc = wmma_acc(a, bb, acc);
  }
  // C/D layout -> row-major T in LDS
#pragma unroll
  for (int r = 0; r < 8; ++r)
    X[(r + 8 * (lane >> 4)) * 16 + (lane & 15)] = acc[r];

  // step 2: Y = T * M
  v8f acc2 = {0, 0, 0, 0, 0, 0, 0, 0};
#pragma unroll
  for (int c = 0; c < 4; ++c) {
    int kb = 4 * c + koff;
    v2f a, bb;
    a.x  = X[mA * 16 + kb + 0];       // A[m][k] = T[m][k]  (adjacent pair)
    a.y  = X[mA * 16 + kb + 1];
    bb.x = M16[(kb + 0) * 16 + nB];   // B[k][n] = M[k][n]
    bb.y = M16[(kb + 1) * 16 + nB];
    acc2 = wmma_acc(a, bb, acc2);
  }

  float* dst = (ch == 0) ? yc : ((ch == 1) ? uc : vc);
  int BW = (ch == 0) ? 64 : 32;
  int bh0 = th * 2, bw0 = tw * 2;
#pragma unroll
  for (int r = 0; r < 8; ++r) {
    int m = r + 8 * (lane >> 4), n = lane & 15;
    int bh = bh0 + (m >> 3), bw = bw0 + (n >> 3);
    int kidx = ((m & 7) << 3) | (n & 7);
    dst[((((size_t)b * BW + bh) * BW + bw) << 6) + kidx] = acc2[r];
  }
}

// ---------------------------------------------------------------------------
// Coefficient-domain init: negC = Q*round(Ws/Q); By = negC; u = 0
// ---------------------------------------------------------------------------
__global__ void k_init_coef(const float* __restrict__ ws, float* __restrict__ negC,
                            float* __restrict__ By, float* __restrict__ Uu, int n) {
  int i = blockIdx.x * blockDim.x + threadIdx.x;
  if (i >= n) return;
  float q = c_qt[i & 63];
  float nc = q * rintf(ws[i] / q);
  negC[i] = nc; By[i] = nc; Uu[i] = 0.0f;
}

// ---------------------------------------------------------------------------
// Fused elementwise coefficient update:
//   AxplusC = (1-a)*By - (2-a)*negC ; r = -AxplusC - Wx - u
//   Wdz = quant_adjust(Wx, r) ; By' = Q*round((Wx + C r + Wdz)/Q)
//   u' = u + AxplusC + By'
// ---------------------------------------------------------------------------
__global__ void k_coef_update(const float* __restrict__ Wx, const float* __restrict__ negC,
                              float* __restrict__ By, float* __restrict__ Uu,
                              float* __restrict__ Rr, float* __restrict__ Wdz, int n) {
  int i = blockIdx.x * blockDim.x + threadIdx.x;
  if (i >= n) return;
  const float AL = 1.8f, CC = 0.5f, MUP = 0.5f;   // C = rho/(mu+rho), MUP = mu/(mu+rho)
  float q = c_qt[i & 63];
  float wx = Wx[i], nc = negC[i], by = By[i], uu = Uu[i];
  float axc = (1.0f - AL) * by - (2.0f - AL) * nc;
  float r = -axc - wx - uu;

  float y  = wx + CC * r;
  float ry = q * rintf(y / q);
  float sgn = (r > 0.0f) ? 1.0f : ((r < 0.0f) ? -1.0f : 0.0f);
  float hqm = 0.5f * q - (1.0f / 510.0f);
  float hqp = 0.5f * q + (1.0f / 510.0f);
  float car = CC * fabsf(r);
  float c1 = -sgn * fminf(car, fabsf(ry - y - sgn * q * hqm));
  float t2 = y - ry - CC * r;
  float sg2 = (t2 > 0.0f) ? 1.0f : ((t2 < 0.0f) ? -1.0f : 0.0f);
  float c2 = ry - hqp * q * sg2 - y;
  float c3 = -sgn * fminf(car, fabsf(ry - y - sgn * q * (1.0f + hqm)));
  auto err = [&](float d) {
    float a1 = CC * r + d;
    float a2 = -MUP + q * rintf((y + d) / q) - y;
    return 0.5f * a1 * a1 + 0.5f * a2 * a2;       // mu/2 = rho/2 = 0.5
  };
  float e1 = err(c1), e2 = err(c2);
  float best = (e2 < e1) ? c2 : c1;
  float eb = (e2 < e1) ? e2 : e1;
  float wdz = (err(c3) < eb) ? c3 : best;

  float bynew = q * rintf((wx + CC * r + wdz) / q);
  Rr[i] = r; Wdz[i] = wdz;
  By[i] = bynew;
  Uu[i] = uu + axc + bynew;
}

// ---------------------------------------------------------------------------
// Fused y-update: per 32x32 pixel region compute IDCT of r and Wdz for Y/U/V,
// clip, 2x chroma upsample, YUV->RGB (both sets), y = x + C*Wt(r) + Wt(Wdz).
// Waves 0-3: Y subtiles; wave 4: U; wave 5: V (each does both coef sets).
// ---------------------------------------------------------------------------
__global__ __launch_bounds__(256) void k_y_update(
    const float* __restrict__ x,
    const float* __restrict__ rY, const float* __restrict__ rU, const float* __restrict__ rV,
    const float* __restrict__ wY, const float* __restrict__ wU, const float* __restrict__ wV,
    float* __restrict__ yimg) {
  __shared__ float M16[256];
  __shared__ float Twk[6][256];
  __shared__ float Ypl[2][1024];
  __shared__ float Upl[2][256];
  __shared__ float Vpl[2][256];

  int tid = threadIdx.x;
  init_M16(M16, tid);
  __syncthreads();

  int wave = tid >> 5, lane = tid & 31;
  int blk = blockIdx.x;
  int b = blk >> 8;
  int ry0 = ((blk >> 4) & 15) * 32;
  int rx0 = (blk & 15) * 32;

  if (wave < 6) {
    int mA = lane & 15, nB = lane & 15, koff = (lane >> 4) * 2;
    int ch, bh0, bw0, ty = 0, tx = 0;
    if (wave < 4) {
      ch = 0; ty = (wave >> 1) * 16; tx = (wave & 1) * 16;
      bh0 = (ry0 + ty) >> 3; bw0 = (rx0 + tx) >> 3;
    } else {
      ch = wave - 3; bh0 = ry0 >> 4; bw0 = rx0 >> 4;   // chroma: 2x2 blocks of 256x256 plane
    }
    int BW = (ch == 0) ? 64 : 32;
    float* T = Twk[wave];

    for (int s = 0; s < 2; ++s) {        // s=0 -> r,  s=1 -> Wdz
      const float* coef = (ch == 0) ? (s ? wY : rY)
                        : (ch == 1) ? (s ? wU : rU)
                                    : (s ? wV : rV);
      // step 1: T = M * Coef   (B fragments gathered straight from global)
      v8f acc = {0, 0, 0, 0, 0, 0, 0, 0};
#pragma unroll
      for (int c = 0; c < 4; ++c) {
        int kb = 4 * c + koff;
        v2f a, bb;
        a.x = M16[mA * 16 + kb + 0];     // A[m][k] = M[m][k]  (adjacent pair)
        a.y = M16[mA * 16 + kb + 1];
        int bw = bw0 + (nB >> 3);
        int k0 = (((kb + 0) & 7) << 3) | (nB & 7);
        int k1 = (((kb + 1) & 7) << 3) | (nB & 7);
        bb.x = coef[((((size_t)b * BW + (bh0 + ((kb + 0) >> 3))) * BW + bw) << 6) + k0];
        bb.y = coef[((((size_t)b * BW + (bh0 + ((kb + 1) >> 3))) * BW + bw) << 6) + k1];
        acc = wmma_acc(a, bb, acc);
      }
#pragma unroll
      for (int r = 0; r < 8; ++r)
        T[(r + 8 * (lane >> 4)) * 16 + (lane & 15)] = acc[r];

      // step 2: X = T * M^T
      v8f acc2 = {0, 0, 0, 0, 0, 0, 0, 0};
#pragma unroll
      for (int c = 0; c < 4; ++c) {
        int kb = 4 * c + koff;
        v2f a, bb;
        a.x  = T[mA * 16 + kb + 0];      // adjacent pair
        a.y  = T[mA * 16 + kb + 1];
        bb.x = M16[nB * 16 + kb + 0];    // B[k][n] = M[n][k]  (adjacent pair)
        bb.y = M16[nB * 16 + kb + 1];
        acc2 = wmma_acc(a, bb, acc2);
      }
      // clip per _Wt and stash plane in LDS
#pragma unroll
      for (int r = 0; r < 8; ++r) {
        int m = r + 8 * (lane >> 4), n = lane & 15;
        float v = acc2[r];
        if (ch == 0)      Ypl[s][(ty + m) * 32 + (tx + n)] = clampf(v + 0.5f, 0.0f, 1.0f);
        else if (ch == 1) Upl[s][m * 16 + n] = clampf(v, -0.5f, 0.5f);
        else              Vpl[s][m * 16 + n] = clampf(v, -0.5f, 0.5f);
      }
    }
  }
  __syncthreads();

  // Combine: upsample chroma, YUV->RGB + clip (both sets), y = x + 0.5*Wt(r) + Wt(Wdz)
  for (int e = tid; e < 1024; e += 256) {
    int i = e >> 5, j = e & 31;
    int ci = (i >> 1) * 16 + (j >> 1);
    float y_r = Ypl[0][e], y_w = Ypl[1][e];
    float u_r = Upl[0][ci], u_w = Upl[1][ci];
    float v_r = Vpl[0][ci], v_w = Vpl[1][ci];
    float Rr_ = clampf(y_r + 1.13988303f  * v_r, 0.f, 1.f);
    float Gr_ = clampf(y_r - 0.394642334f * u_r - 0.58062185f * v_r, 0.f, 1.f);
    float Br_ = clampf(y_r + 2.03206185f  * u_r, 0.f, 1.f);
    float Rw_ = clampf(y_w + 1.13988303f  * v_w, 0.f, 1.f);
    float Gw_ = clampf(y_w - 0.394642334f * u_w - 0.58062185f * v_w, 0.f, 1.f);
    float Bw_ = clampf(y_w + 2.03206185f  * u_w, 0.f, 1.f);
    size_t base = (((size_t)b * H_ + (ry0 + i)) * W_ + (rx0 + j)) * 3;
    yimg[base + 0] = x[base + 0] + 0.5f * Rr_ + Rw_;
    yimg[base + 1] = x[base + 1] + 0.5f * Gr_ + Gw_;
    yimg[base + 2] = x[base + 2] + 0.5f * Br_ + Bw_;
  }
}

// xup collapses to t/3 (impulse filters => |F1|=|F2|=1 => A = 1+lambda*2 = 3)
__global__ void k_scale(float* __restrict__ dst, const float* __restrict__ src,
                        int n, float s) {
  int i = blockIdx.x * blockDim.x + threadIdx.x;
  if (i < n) dst[i] = src[i] * s;
}

extern "C" void kernel_launch(void* const* d_in, const int* in_sizes, int n_in,
                              void* d_out, int out_size, void* d_ws, size_t ws_size,
                              hipStream_t stream) {
  (void)in_sizes; (void)n_in; (void)out_size; (void)ws_size;
  const float* s = (const float*)d_in[0];
  float* out = (float*)d_out;

  float* p = (float*)d_ws;
  float* y    = p; p += NIMG;
  float* x    = p; p += NIMG;
  float* negC = p; p += NCOEF;
  float* By   = p; p += NCOEF;
  float* Uu   = p; p += NCOEF;
  float* Wx   = p; p += NCOEF;
  float* Rr   = p; p += NCOEF;
  float* Wz   = p; p += NCOEF;   // total ~60 MB

  hipMemcpyAsync(y, s, (size_t)NIMG * sizeof(float), hipMemcpyDeviceToDevice, stream);

  // Ws = W(img); negC = Q*round(Ws/Q); By = negC; u = 0
  k_forward_W<<<768, 256, 0, stream>>>(s, Wx, Wx + NYC, Wx + NYC + NUC);
  k_init_coef<<<(NCOEF + 255) / 256, 256, 0, stream>>>(Wx, negC, By, Uu, NCOEF);

  for (int it = 0; it < 4; ++it) {
    k_scale<<<(NIMG + 255) / 256, 256, 0, stream>>>(x, y, NIMG, 1.0f / 3.0f);
    k_forward_W<<<768, 256, 0, stream>>>(x, Wx, Wx + NYC, Wx + NYC + NUC);
    k_coef_update<<<(NCOEF + 255) / 256, 256, 0, stream>>>(Wx, negC, By, Uu, Rr, Wz, NCOEF);
    k_y_update<<<1024, 256, 0, stream>>>(x,
        Rr, Rr + NYC, Rr + NYC + NUC,
        Wz, Wz + NYC, Wz + NYC + NUC, y);
  }
  k_scale<<<(NIMG + 255) / 256, 256, 0, stream>>>(out, y, NIMG, 1.0f / 3.0f);
}